// MultiheadAttention_72533407694898
// MI455X (gfx1250) — compile-verified
//
#include <hip/hip_runtime.h>
#include <hip/hip_bf16.h>

// ---------------------------------------------------------------------------
// MI455X (gfx1250) multihead-attention variant, fp32 via V_WMMA_F32_16X16X4_F32
// + async global->LDS double-buffered tile streaming (ASYNCcnt path).
// ---------------------------------------------------------------------------

typedef float v2f __attribute__((ext_vector_type(2)));
typedef float v8f __attribute__((ext_vector_type(8)));

#define WMMA4(a, b, c) \
  __builtin_amdgcn_wmma_f32_16x16x4_f32(false, (a), false, (b), (short)0, (c), false, false)

#define WAIT_ASYNC0() asm volatile("s_wait_asynccnt 0x0" ::: "memory")

__device__ __forceinline__ v8f zero8() {
  v8f z = {0.f, 0.f, 0.f, 0.f, 0.f, 0.f, 0.f, 0.f};
  return z;
}

// Problem constants
#define T_LEN   1024
#define BSZ_    16
#define E_DIM   512
#define H_NUM   8
#define HD      64
#define S_LEN   1025
#define S_PAD   1040           // S rounded up to multiple of 16, pad rows zeroed
#define F3E     (3 * E_DIM)    // 1536
#define LDW     68             // LDS row stride (floats): 4-bank skew, 16B aligned

// Output offsets (floats) in d_out: [attn | aw_avg | w_ret]
#define OFS_AVG  (T_LEN * BSZ_ * E_DIM)                   // 8,388,608
#define OFS_WRET (OFS_AVG + BSZ_ * T_LEN * S_LEN)          // 25,182,208

// Workspace offsets (floats)
#define WS_W    0
#define WS_Q    (WS_W + F3E * E_DIM)
#define WS_K    (WS_Q + 128 * T_LEN * HD)
#define WS_V    (WS_K + 128 * S_PAD * HD)
#define WS_ATT  (WS_V + 128 * S_PAD * HD)
#define WS_PART (WS_ATT + 128 * T_LEN * HD)
#define WS_SCAL (WS_PART + 2 * 2048)

// ---------------------------------------------------------------------------
// Async copy of one 16x64 fp32 tile (global, row stride 64) into LDS with row
// stride LDW. 256 x b128 transfers issued by 128 threads; tracked by ASYNCcnt.
// LDS offset = low 32 bits of the flat shared pointer (ISA 10.2 aperture map).
// ---------------------------------------------------------------------------
__device__ __forceinline__ void cp_tile_async(const float* __restrict__ g,
                                              float* l, int tid) {
#pragma unroll
  for (int j = 0; j < 2; ++j) {
    int t = tid + j * 128;
    int row = t >> 4, c4 = (t & 15) << 2;
    unsigned long long ga = (unsigned long long)(uintptr_t)(g + row * HD + c4);
    unsigned int la = (unsigned int)(uintptr_t)(l + row * LDW + c4);
    asm volatile("global_load_async_to_lds_b128 %0, %1, off"
                 :: "v"(la), "v"(ga) : "memory");
  }
}

// ---------------------------------------------------------------------------
// Kernel 1: build mutated weight w (every 3rd column <- i_proj) and w_ret out.
// ---------------------------------------------------------------------------
__global__ void prep_w(const float* __restrict__ Win, const float* __restrict__ iproj,
                       float* __restrict__ wmut, float* __restrict__ wret) {
  int idx = blockIdx.x * 256 + threadIdx.x;   // < 1536*512
  int f = idx >> 9, e = idx & 511;
  float val = ((e % 3) == 0) ? iproj[f] : Win[idx];
  wmut[idx] = val;
  int c = f >> 9, f2 = f & 511;               // w_ret[c, e, f2] = w[c*512+f2, e]
  wret[c * (E_DIM * E_DIM) + e * E_DIM + f2] = val;
}

// ---------------------------------------------------------------------------
// Kernel 2: qkv = X(16384x512) @ w^T(512x1536) + bias; scatter to head layout.
// 256 threads = 8 waves; each wave owns a 16x16 output tile.
// ---------------------------------------------------------------------------
__global__ void qkv_gemm(const float* __restrict__ X, const float* __restrict__ W,
                         const float* __restrict__ bias,
                         float* __restrict__ qb, float* __restrict__ kb,
                         float* __restrict__ vb) {
  int lane = threadIdx.x & 31, lo = lane & 15, hi = lane >> 4;
  int w = threadIdx.x >> 5;
  int m0 = (blockIdx.x * 8 + w) * 16;
  int f0 = blockIdx.y * 16;
  v8f acc = zero8();
  for (int k = 0; k < E_DIM; k += 4) {
    int kk = k + 2 * hi;
    const float* ap = X + (m0 + lo) * E_DIM + kk;       // A[m][k]
    const float* bp = W + (f0 + lo) * E_DIM + kk;       // B[k][f] = W[f][k]
    v2f a = {ap[0], ap[1]};
    v2f b = {bp[0], bp[1]};
    acc = WMMA4(a, b, acc);
  }
  int F = f0 + lo;
  int sec = F >> 9, e = F & 511, h = e >> 6, d = e & 63;
  float bia = bias[F];
#pragma unroll
  for (int r = 0; r < 8; ++r) {
    int R = m0 + r + 8 * hi;                             // flat row = t*16 + b
    int t = R >> 4, b_ = R & 15;
    int nh = b_ * H_NUM + h;
    float val = acc[r] + bia;
    if (sec == 0)      qb[(nh * T_LEN + t) * HD + d] = val * 0.125f; // hd^-0.5
    else if (sec == 1) kb[(nh * S_PAD + t) * HD + d] = val;
    else               vb[(nh * S_PAD + t) * HD + d] = val;
  }
}

// ---------------------------------------------------------------------------
// Kernel 3: append bias_k/bias_v row (s=1024) and zero pad rows 1025..1039.
// ---------------------------------------------------------------------------
__global__ void pad_kv(const float* __restrict__ bk, const float* __restrict__ bv,
                       float* __restrict__ k, float* __restrict__ v) {
  int idx = blockIdx.x * 256 + threadIdx.x;   // 128 heads * 64 dims
  int n = idx >> 6, d = idx & 63, h = n & 7;
  float* kp = k + (n * S_PAD + T_LEN) * HD + d;
  float* vp = v + (n * S_PAD + T_LEN) * HD + d;
  kp[0] = bk[h * HD + d];
  vp[0] = bv[h * HD + d];
#pragma unroll
  for (int r = 1; r < 16; ++r) { kp[r * HD] = 0.f; vp[r * HD] = 0.f; }
}

// ---------------------------------------------------------------------------
// Kernel 4: pass 1 over scores — per-block min/max partials.
// grid(16 t-blocks, 128 heads), 128 threads = 4 waves, wave = 16 t-rows.
// K tile shared by the 4 waves via double-buffered async LDS staging.
// ---------------------------------------------------------------------------
__global__ void score_minmax(const float* __restrict__ q, const float* __restrict__ k,
                             float* __restrict__ part) {
  __shared__ float kbuf[2][16 * LDW];
  __shared__ float smn[128], smx[128];
  int tid = threadIdx.x;
  int lane = tid & 31, lo = lane & 15, hi = lane >> 4, w = tid >> 5;
  int n = blockIdx.y;
  int t0 = blockIdx.x * 64 + w * 16;
  const float* qh = q + (n * T_LEN + t0) * HD;
  const float* kh = k + n * S_PAD * HD;
  // s-invariant A (q) fragments held in registers
  v2f aq[16];
#pragma unroll
  for (int i = 0; i < 16; ++i) {
    int kc = i * 4 + 2 * hi;
    aq[i] = v2f{qh[lo * HD + kc], qh[lo * HD + kc + 1]};
  }
  float mn = 3.4e38f, mx = -3.4e38f;
  cp_tile_async(kh, kbuf[0], tid);
  int pb = 0;
  for (int s0 = 0; s0 < S_PAD; s0 += 16) {
    WAIT_ASYNC0();
    __syncthreads();                            // tile pb ready for all waves
    if (s0 + 16 < S_PAD) cp_tile_async(kh + (s0 + 16) * HD, kbuf[pb ^ 1], tid);
    const float* kt = kbuf[pb];
    v8f acc = zero8();
#pragma unroll
    for (int i = 0; i < 16; ++i) {
      int kc = i * 4 + 2 * hi;
      v2f b = {kt[lo * LDW + kc], kt[lo * LDW + kc + 1]};  // B[k][s] = k[s][k]
      acc = WMMA4(aq[i], b, acc);
    }
    if (s0 + lo < S_LEN) {                      // mask pad columns
#pragma unroll
      for (int r = 0; r < 8; ++r) {
        float x = acc[r];
        mn = fminf(mn, x);
        mx = fmaxf(mx, x);
      }
    }
    pb ^= 1;
  }
  smn[tid] = mn;
  smx[tid] = mx;
  __syncthreads();
  for (int off = 64; off > 0; off >>= 1) {
    if (tid < off) {
      smn[tid] = fminf(smn[tid], smn[tid + off]);
      smx[tid] = fmaxf(smx[tid], smx[tid + off]);
    }
    __syncthreads();
  }
  if (tid == 0) {
    int bid = blockIdx.y * gridDim.x + blockIdx.x;
    part[2 * bid] = smn[0];
    part[2 * bid + 1] = smx[0];
  }
}

// ---------------------------------------------------------------------------
// Kernel 5: reduce 2048 partials -> scal[0]=mn, scal[1]=1/(mx-mn).
// ---------------------------------------------------------------------------
__global__ void reduce_minmax(const float* __restrict__ part, int npart,
                              float* __restrict__ scal) {
  __shared__ float smn[256], smx[256];
  float mn = 3.4e38f, mx = -3.4e38f;
  for (int i = threadIdx.x; i < npart; i += 256) {
    mn = fminf(mn, part[2 * i]);
    mx = fmaxf(mx, part[2 * i + 1]);
  }
  smn[threadIdx.x] = mn;
  smx[threadIdx.x] = mx;
  __syncthreads();
  for (int off = 128; off > 0; off >>= 1) {
    if ((int)threadIdx.x < off) {
      smn[threadIdx.x] = fminf(smn[threadIdx.x], smn[threadIdx.x + off]);
      smx[threadIdx.x] = fmaxf(smx[threadIdx.x], smx[threadIdx.x + off]);
    }
    __syncthreads();
  }
  if (threadIdx.x == 0) {
    scal[0] = smn[0];
    scal[1] = 1.0f / (smx[0] - smn[0]);
  }
}

// ---------------------------------------------------------------------------
// Kernel 6: pass 2 — recompute scores, normalize, attn = awn @ v.
// K and V tiles double-buffered through async LDS staging; awn restaged
// C-layout -> A-layout via a wave-private LDS tile.
// grid(16 t-blocks, 128 heads), 128 threads = 4 waves.
// ---------------------------------------------------------------------------
__global__ void attn_v(const float* __restrict__ q, const float* __restrict__ k,
                       const float* __restrict__ v, const float* __restrict__ scal,
                       float* __restrict__ att) {
  __shared__ float kbuf[2][16 * LDW];
  __shared__ float vbuf[2][16 * LDW];
  __shared__ float awn[4][16 * 17];
  int tid = threadIdx.x;
  int lane = tid & 31, lo = lane & 15, hi = lane >> 4, w = tid >> 5;
  float* tile = awn[w];
  int n = blockIdx.y;
  int t0 = blockIdx.x * 64 + w * 16;
  const float* qh = q + (n * T_LEN + t0) * HD;
  const float* kh = k + n * S_PAD * HD;
  const float* vh = v + n * S_PAD * HD;
  float mn = scal[0], inv = scal[1];
  v2f aq[16];
#pragma unroll
  for (int i = 0; i < 16; ++i) {
    int kc = i * 4 + 2 * hi;
    aq[i] = v2f{qh[lo * HD + kc], qh[lo * HD + kc + 1]};
  }
  v8f accd[4] = {zero8(), zero8(), zero8(), zero8()};   // 16 rows x 64 cols
  cp_tile_async(kh, kbuf[0], tid);
  cp_tile_async(vh, vbuf[0], tid);
  int pb = 0;
  for (int s0 = 0; s0 < S_PAD; s0 += 16) {
    WAIT_ASYNC0();
    __syncthreads();
    if (s0 + 16 < S_PAD) {
      cp_tile_async(kh + (s0 + 16) * HD, kbuf[pb ^ 1], tid);
      cp_tile_async(vh + (s0 + 16) * HD, vbuf[pb ^ 1], tid);
    }
    const float* kt = kbuf[pb];
    const float* vt = vbuf[pb];
    v8f aw = zero8();
#pragma unroll
    for (int i = 0; i < 16; ++i) {
      int kc = i * 4 + 2 * hi;
      v2f b = {kt[lo * LDW + kc], kt[lo * LDW + kc + 1]};
      aw = WMMA4(aq[i], b, aw);
    }
    // normalize + stage to LDS in (row,col) form; pad columns hit zeroed v rows
#pragma unroll
    for (int r = 0; r < 8; ++r)
      tile[(r + 8 * hi) * 17 + lo] = (aw[r] - mn) * inv;
    // awn(16x16) @ v(16x64): A from wave-private LDS, B from staged v tile
#pragma unroll
    for (int c = 0; c < 4; ++c) {
      int d0 = c * 16;
#pragma unroll
      for (int kk = 0; kk < 16; kk += 4) {
        int kc = kk + 2 * hi;
        v2f a = {tile[lo * 17 + kc], tile[lo * 17 + kc + 1]};
        v2f b = {vt[kc * LDW + d0 + lo], vt[(kc + 1) * LDW + d0 + lo]};
        accd[c] = WMMA4(a, b, accd[c]);
      }
    }
    pb ^= 1;
  }
  float* ah = att + (n * T_LEN + t0) * HD;
#pragma unroll
  for (int c = 0; c < 4; ++c)
#pragma unroll
    for (int r = 0; r < 8; ++r)
      ah[(r + 8 * hi) * HD + c * 16 + lo] = accd[c][r];
}

// ---------------------------------------------------------------------------
// Kernel 7: aw_avg via algebraic identity: sum_h aw_h = full-E dot(q_scaled,k).
// avg = (dot - 8*mn) * inv / 8.  grid(16 t-blocks, 16 batches), 128 threads.
// ---------------------------------------------------------------------------
__global__ void avg_scores(const float* __restrict__ q, const float* __restrict__ k,
                           const float* __restrict__ scal, float* __restrict__ avg) {
  int lane = threadIdx.x & 31, lo = lane & 15, hi = lane >> 4, w = threadIdx.x >> 5;
  int b = blockIdx.y;
  int t0 = blockIdx.x * 64 + w * 16;
  float mn = scal[0], inv = scal[1];
  for (int s0 = 0; s0 < S_PAD; s0 += 16) {
    v8f acc = zero8();
    for (int e = 0; e < E_DIM; e += 4) {
      int ee = e + 2 * hi;                    // even -> (h,d) with d even
      int h = ee >> 6, d = ee & 63;
      const float* qh = q + ((b * H_NUM + h) * T_LEN + t0 + lo) * HD + d;
      const float* kh = k + ((b * H_NUM + h) * S_PAD + s0 + lo) * HD + d;
      v2f a = {qh[0], qh[1]};
      v2f bb = {kh[0], kh[1]};
      acc = WMMA4(a, bb, acc);
    }
    int s = s0 + lo;
    if (s < S_LEN) {
#pragma unroll
      for (int r = 0; r < 8; ++r) {
        int t = t0 + r + 8 * hi;
        avg[(b * T_LEN + t) * S_LEN + s] = (acc[r] - 8.0f * mn) * inv * 0.125f;
      }
    }
  }
}

// ---------------------------------------------------------------------------
// Kernel 8: out-proj: attn(16384x512) @ out_w^T(512x512) + out_b.
// 256 threads = 8 waves; wave = 16x16 tile.
// ---------------------------------------------------------------------------
__global__ void out_proj(const float* __restrict__ att, const float* __restrict__ Wo,
                         const float* __restrict__ bo, float* __restrict__ out) {
  int lane = threadIdx.x & 31, lo = lane & 15, hi = lane >> 4, w = threadIdx.x >> 5;
  int m0 = (blockIdx.x * 8 + w) * 16;
  int n0 = blockIdx.y * 16;
  int R = m0 + lo, t = R >> 4, b_ = R & 15;
  v8f acc = zero8();
  for (int e = 0; e < E_DIM; e += 4) {
    int ee = e + 2 * hi;
    int h = ee >> 6, d = ee & 63;
    const float* ap = att + ((b_ * H_NUM + h) * T_LEN + t) * HD + d;
    const float* bp = Wo + (n0 + lo) * E_DIM + ee;   // B[e][eo] = out_w[eo][e]
    v2f a = {ap[0], ap[1]};
    v2f b = {bp[0], bp[1]};
    acc = WMMA4(a, b, acc);
  }
  float bia = bo[n0 + lo];
#pragma unroll
  for (int r = 0; r < 8; ++r)
    out[(m0 + r + 8 * hi) * E_DIM + n0 + lo] = acc[r] + bia;
}

// ---------------------------------------------------------------------------
extern "C" void kernel_launch(void* const* d_in, const int* in_sizes, int n_in,
                              void* d_out, int out_size, void* d_ws, size_t ws_size,
                              hipStream_t stream) {
  const float* query = (const float*)d_in[0];
  const float* iproj = (const float*)d_in[1];
  const float* winp  = (const float*)d_in[2];
  const float* binp  = (const float*)d_in[3];
  const float* outw  = (const float*)d_in[4];
  const float* outb  = (const float*)d_in[5];
  const float* biask = (const float*)d_in[6];
  const float* biasv = (const float*)d_in[7];
  float* out = (float*)d_out;
  float* ws  = (float*)d_ws;

  float* w_mut = ws + WS_W;
  float* Q = ws + WS_Q;
  float* K = ws + WS_K;
  float* V = ws + WS_V;
  float* ATT = ws + WS_ATT;
  float* PART = ws + WS_PART;
  float* SCAL = ws + WS_SCAL;

  prep_w<<<3072, 256, 0, stream>>>(winp, iproj, w_mut, out + OFS_WRET);
  qkv_gemm<<<dim3(128, 96), 256, 0, stream>>>(query, w_mut, binp, Q, K, V);
  pad_kv<<<32, 256, 0, stream>>>(biask, biasv, K, V);
  score_minmax<<<dim3(16, 128), 128, 0, stream>>>(Q, K, PART);
  reduce_minmax<<<1, 256, 0, stream>>>(PART, 2048, SCAL);
  attn_v<<<dim3(16, 128), 128, 0, stream>>>(Q, K, V, SCAL, ATT);
  avg_scores<<<dim3(16, 16), 128, 0, stream>>>(Q, K, SCAL, out + OFS_AVG);
  out_proj<<<dim3(128, 32), 256, 0, stream>>>(ATT, outw, outb, out);
}